// ConvModel_35304631173416
// MI455X (gfx1250) — compile-verified
//
#include <hip/hip_runtime.h>

// ---------------------------------------------------------------------------
// CDNA5 f32 WMMA: D[16x16] = A[16x4] * B[4x16] + C, wave32.
// A frag (2 VGPR):  lane L -> M = L%16, v0 = A[M][k0+2*(L/16)], v1 = +1
// B frag (2 VGPR):  lane L -> N = L%16, v0 = B[k0+2*(L/16)][N], v1 = +1
// C/D    (8 VGPR):  VGPR r -> row M=r (lanes 0-15), M=r+8 (lanes 16-31), N=L%16
// ---------------------------------------------------------------------------
typedef float v2f __attribute__((ext_vector_type(2)));
typedef float v8f __attribute__((ext_vector_type(8)));

__device__ __forceinline__ v8f wmma_f32(v2f a, v2f b, v8f c) {
  return __builtin_amdgcn_wmma_f32_16x16x4_f32(
      /*neg_a=*/false, a, /*neg_b=*/false, b,
      /*c_mod=*/(short)0, c, /*reuse_a=*/false, /*reuse_b=*/false);
}

#define PAD_DE 68    // 64 + 4 floats  (even stride, breaks 64-bank periodicity)
#define PAD_K 260    // 256 + 4 floats
#define EDGE_WAVES 8
#define NODE_WAVES 4

// ---------------------------------------------------------------------------
__global__ void zero_f32(float* __restrict__ p, size_t n) {
  size_t i = (size_t)blockIdx.x * blockDim.x + threadIdx.x;
  if (i < n) p[i] = 0.f;
}

__global__ void deg_kernel(const int* __restrict__ dst, float* __restrict__ deg, int E) {
  int i = blockIdx.x * blockDim.x + threadIdx.x;
  if (i < E) atomicAdd(&deg[dst[i]], 1.0f);
}

// agg[dst[e], 0:128] += h_neigh[src[e], :]  (32 lanes per edge, float4 gather)
__global__ void neigh_scatter(const float* __restrict__ h_neigh,
                              const int* __restrict__ src,
                              const int* __restrict__ dst,
                              float* __restrict__ agg, int E) {
  long long gid = (long long)blockIdx.x * blockDim.x + threadIdx.x;
  long long e = gid >> 5;
  int lane = (int)(gid & 31);
  if (e >= E) return;
  int s = src[e], d = dst[e];
  float4 v = *(const float4*)(h_neigh + (long long)s * 128 + lane * 4);
  float* p = agg + (long long)d * 256 + lane * 4;
  atomicAdd(p + 0, v.x);
  atomicAdd(p + 1, v.y);
  atomicAdd(p + 2, v.z);
  atomicAdd(p + 3, v.w);
}

// ---------------------------------------------------------------------------
// Edge GEMM: eh = l2norm(relu(e_feats[16-tile] @ W_edge^T)); atomic-scatter
// into agg[dst, 128:256]. One 16-edge tile per wave, 8 waves / block.
// ---------------------------------------------------------------------------
__global__ __launch_bounds__(256) void edge_gemm_scatter(
    const float* __restrict__ e_feats, const float* __restrict__ W_edge,
    const int* __restrict__ dst_idx, float* __restrict__ agg, int E) {
  extern __shared__ float smem[];
  float* Wl = smem;                                   // [128][PAD_DE]
  int tid = threadIdx.x;
  int wave = tid >> 5, lane = tid & 31;
  int m = lane & 15, half = lane >> 4;
  float* Aw = smem + 128 * PAD_DE + wave * 16 * PAD_DE;  // [16][PAD_DE]

  // stage W_edge (128x64) padded
  for (int i = tid; i < 128 * 16; i += 256) {
    int row = i >> 4, c4 = i & 15;
    float4 v = ((const float4*)W_edge)[i];
    *(float4*)&Wl[row * PAD_DE + c4 * 4] = v;
  }
  // stage e_feats tile (16x64), zero-fill OOB rows
  long long tile = (long long)blockIdx.x * EDGE_WAVES + wave;
  long long base = tile * 16;
  for (int i = lane; i < 16 * 16; i += 32) {
    int r = i >> 4, c4 = i & 15;
    long long e = base + r;
    float4 v = make_float4(0.f, 0.f, 0.f, 0.f);
    if (e < E) v = ((const float4*)(e_feats + e * 64))[c4];
    *(float4*)&Aw[r * PAD_DE + c4 * 4] = v;
  }
  __syncthreads();
  if (base >= E) return;

  v8f acc[8] = {};
  const float* aRow = &Aw[m * PAD_DE];
  for (int k0 = 0; k0 < 64; k0 += 4) {
    v2f a = *(const v2f*)&aRow[k0 + 2 * half];
#pragma unroll
    for (int nt = 0; nt < 8; ++nt) {
      v2f b = *(const v2f*)&Wl[(nt * 16 + m) * PAD_DE + k0 + 2 * half];
      acc[nt] = wmma_f32(a, b, acc[nt]);
    }
  }

  // relu + per-row L2 norm (butterfly within each 16-lane half)
  float nrm[8];
#pragma unroll
  for (int r = 0; r < 8; ++r) {
    float s = 0.f;
#pragma unroll
    for (int nt = 0; nt < 8; ++nt) {
      float v = fmaxf(acc[nt][r], 0.f);
      acc[nt][r] = v;
      s += v * v;
    }
    nrm[r] = s;
  }
#pragma unroll
  for (int off = 1; off < 16; off <<= 1) {
#pragma unroll
    for (int r = 0; r < 8; ++r) nrm[r] += __shfl_xor(nrm[r], off, 32);
  }
#pragma unroll
  for (int r = 0; r < 8; ++r) nrm[r] = (nrm[r] > 0.f) ? (1.0f / sqrtf(nrm[r])) : 1.0f;

  // atomic scatter: row M = r + 8*half of this tile -> agg[dst, 128 + n]
#pragma unroll
  for (int r = 0; r < 8; ++r) {
    long long e = base + r + 8 * half;
    if (e < E) {
      int d = dst_idx[e];
      float* p = agg + (long long)d * 256 + 128;
      float sc = nrm[r];
#pragma unroll
      for (int nt = 0; nt < 8; ++nt) atomicAdd(&p[nt * 16 + m], acc[nt][r] * sc);
    }
  }
}

// ---------------------------------------------------------------------------
// hn = relu((agg/deg) @ W_remap^T)   [16-node tile per wave]
// ---------------------------------------------------------------------------
__global__ __launch_bounds__(128) void remap_gemm(
    const float* __restrict__ agg, const float* __restrict__ deg,
    const float* __restrict__ W_remap, float* __restrict__ hn, int NDST) {
  extern __shared__ float smem[];
  float* Wl = smem;                                  // [128][PAD_K]
  int tid = threadIdx.x;
  int wave = tid >> 5, lane = tid & 31;
  int m = lane & 15, half = lane >> 4;
  float* Aw = smem + 128 * PAD_K + wave * 16 * PAD_K;  // [16][PAD_K]

  for (int i = tid; i < 128 * 64; i += 128) {   // 128x256 floats = 8192 float4
    int row = i >> 6, c4 = i & 63;
    float4 v = ((const float4*)W_remap)[i];
    *(float4*)&Wl[row * PAD_K + c4 * 4] = v;
  }
  long long tile = (long long)blockIdx.x * NODE_WAVES + wave;
  long long nbase = tile * 16;
  float invd = 1.f;
  if (lane < 16 && nbase + lane < NDST) invd = 1.0f / fmaxf(deg[nbase + lane], 1.0f);
  for (int i = lane; i < 16 * 64; i += 32) {    // per iter, row r is wave-uniform
    int r = i >> 6, c4 = i & 63;
    long long nd = nbase + r;
    float4 v = make_float4(0.f, 0.f, 0.f, 0.f);
    if (nd < NDST) {
      v = ((const float4*)(agg + nd * 256))[c4];
      float s = __shfl(invd, r, 32);
      v.x *= s; v.y *= s; v.z *= s; v.w *= s;
    }
    *(float4*)&Aw[r * PAD_K + c4 * 4] = v;
  }
  __syncthreads();
  if (nbase >= NDST) return;

  v8f acc[8] = {};
  const float* aRow = &Aw[m * PAD_K];
  for (int k0 = 0; k0 < 256; k0 += 4) {
    v2f a = *(const v2f*)&aRow[k0 + 2 * half];
#pragma unroll
    for (int nt = 0; nt < 8; ++nt) {
      v2f b = *(const v2f*)&Wl[(nt * 16 + m) * PAD_K + k0 + 2 * half];
      acc[nt] = wmma_f32(a, b, acc[nt]);
    }
  }
#pragma unroll
  for (int r = 0; r < 8; ++r) {
    long long nd = nbase + r + 8 * half;
    if (nd < NDST) {
      float* p = hn + nd * 128;
#pragma unroll
      for (int nt = 0; nt < 8; ++nt) p[nt * 16 + m] = fmaxf(acc[nt][r], 0.f);
    }
  }
}

// ---------------------------------------------------------------------------
// out = l2norm(relu(h_self@W_self^T + hn@W_neigh^T))
// as one K=256 GEMM: A = [h_self | hn], B = [W_self^T ; W_neigh^T]
// ---------------------------------------------------------------------------
__global__ __launch_bounds__(128) void out_gemm(
    const float* __restrict__ h_self, const float* __restrict__ hn,
    const float* __restrict__ W_self, const float* __restrict__ W_neigh,
    float* __restrict__ out, int NDST) {
  extern __shared__ float smem[];
  float* Wl = smem;                                  // [128][PAD_K] concat cols
  int tid = threadIdx.x;
  int wave = tid >> 5, lane = tid & 31;
  int m = lane & 15, half = lane >> 4;
  float* Aw = smem + 128 * PAD_K + wave * 16 * PAD_K;

  for (int i = tid; i < 128 * 32; i += 128) {   // each matrix: 4096 float4
    int row = i >> 5, c4 = i & 31;
    *(float4*)&Wl[row * PAD_K + c4 * 4] = ((const float4*)W_self)[i];
    *(float4*)&Wl[row * PAD_K + 128 + c4 * 4] = ((const float4*)W_neigh)[i];
  }
  long long tile = (long long)blockIdx.x * NODE_WAVES + wave;
  long long nbase = tile * 16;
  for (int i = lane; i < 16 * 32; i += 32) {
    int r = i >> 5, c4 = i & 31;
    long long nd = nbase + r;
    float4 a = make_float4(0.f, 0.f, 0.f, 0.f);
    float4 b = make_float4(0.f, 0.f, 0.f, 0.f);
    if (nd < NDST) {
      a = ((const float4*)(h_self + nd * 128))[c4];
      b = ((const float4*)(hn + nd * 128))[c4];
    }
    *(float4*)&Aw[r * PAD_K + c4 * 4] = a;
    *(float4*)&Aw[r * PAD_K + 128 + c4 * 4] = b;
  }
  __syncthreads();
  if (nbase >= NDST) return;

  v8f acc[8] = {};
  const float* aRow = &Aw[m * PAD_K];
  for (int k0 = 0; k0 < 256; k0 += 4) {
    v2f a = *(const v2f*)&aRow[k0 + 2 * half];
#pragma unroll
    for (int nt = 0; nt < 8; ++nt) {
      v2f b = *(const v2f*)&Wl[(nt * 16 + m) * PAD_K + k0 + 2 * half];
      acc[nt] = wmma_f32(a, b, acc[nt]);
    }
  }
  float nrm[8];
#pragma unroll
  for (int r = 0; r < 8; ++r) {
    float s = 0.f;
#pragma unroll
    for (int nt = 0; nt < 8; ++nt) {
      float v = fmaxf(acc[nt][r], 0.f);
      acc[nt][r] = v;
      s += v * v;
    }
    nrm[r] = s;
  }
#pragma unroll
  for (int off = 1; off < 16; off <<= 1) {
#pragma unroll
    for (int r = 0; r < 8; ++r) nrm[r] += __shfl_xor(nrm[r], off, 32);
  }
#pragma unroll
  for (int r = 0; r < 8; ++r) nrm[r] = (nrm[r] > 0.f) ? (1.0f / sqrtf(nrm[r])) : 1.0f;
#pragma unroll
  for (int r = 0; r < 8; ++r) {
    long long nd = nbase + r + 8 * half;
    if (nd < NDST) {
      float* p = out + nd * 128;
      float sc = nrm[r];
#pragma unroll
      for (int nt = 0; nt < 8; ++nt) p[nt * 16 + m] = acc[nt][r] * sc;
    }
  }
}

// ---------------------------------------------------------------------------
extern "C" void kernel_launch(void* const* d_in, const int* in_sizes, int n_in,
                              void* d_out, int out_size, void* d_ws, size_t ws_size,
                              hipStream_t stream) {
  const float* h_neigh = (const float*)d_in[0];
  const float* h_self  = (const float*)d_in[1];
  const float* e_feats = (const float*)d_in[2];
  const float* W_edge  = (const float*)d_in[3];
  const float* W_remap = (const float*)d_in[4];
  const float* W_self  = (const float*)d_in[5];
  const float* W_neigh = (const float*)d_in[6];
  const int* src_idx = (const int*)d_in[7];
  const int* dst_idx = (const int*)d_in[8];

  const int NDST = in_sizes[1] / 128;
  const int E = in_sizes[7];

  // workspace: agg [NDST,256] | deg [NDST] | hn [NDST,128]
  size_t need = ((size_t)NDST * 256 + NDST + (size_t)NDST * 128) * sizeof(float);
  if (ws_size < need) return;
  float* agg = (float*)d_ws;
  float* deg = agg + (size_t)NDST * 256;
  float* hn  = deg + NDST;
  float* out = (float*)d_out;

  size_t nz = (size_t)NDST * 257;  // agg + deg are contiguous
  zero_f32<<<(unsigned)((nz + 1023) / 1024), 1024, 0, stream>>>(agg, nz);
  deg_kernel<<<(E + 255) / 256, 256, 0, stream>>>(dst_idx, deg, E);

  long long nth = (long long)E * 32;
  neigh_scatter<<<(unsigned)((nth + 255) / 256), 256, 0, stream>>>(h_neigh, src_idx,
                                                                   dst_idx, agg, E);

  int etiles = (E + 15) / 16;
  size_t esmem = (size_t)(128 * PAD_DE + EDGE_WAVES * 16 * PAD_DE) * sizeof(float);
  edge_gemm_scatter<<<(etiles + EDGE_WAVES - 1) / EDGE_WAVES, 256, esmem, stream>>>(
      e_feats, W_edge, dst_idx, agg, E);

  int ntiles = (NDST + 15) / 16;
  size_t nsmem = (size_t)(128 * PAD_K + NODE_WAVES * 16 * PAD_K) * sizeof(float);
  remap_gemm<<<(ntiles + NODE_WAVES - 1) / NODE_WAVES, 128, nsmem, stream>>>(
      agg, deg, W_remap, hn, NDST);
  out_gemm<<<(ntiles + NODE_WAVES - 1) / NODE_WAVES, 128, nsmem, stream>>>(
      h_self, hn, W_self, W_neigh, out, NDST);
}